// CTCLoss_72009421684846
// MI455X (gfx1250) — compile-verified
//
#include <hip/hip_runtime.h>
#include <hip/hip_bf16.h>
#include <stdint.h>
#include <math.h>

#define B_ 64
#define C_ 128
#define T_ 512
#define L_ 128
#define S_ 257
#define NTHR 288          // 9 wave32s, covers S=257 states
#define NWAVE 9

typedef __attribute__((ext_vector_type(2))) float v2f;
typedef __attribute__((ext_vector_type(8))) float v8f;

// ---- CDNA5 async global->LDS copy (ASYNCcnt-tracked) -----------------------
__device__ __forceinline__ void async_copy_b32(void* lds_dst, const float* gsrc) {
    unsigned lds = (unsigned)(uintptr_t)lds_dst;   // low 32 bits of generic = LDS offset
    asm volatile("global_load_async_to_lds_b32 %0, %1, off"
                 :: "v"(lds), "v"(gsrc) : "memory");
}
__device__ __forceinline__ void wait_async_le1() {
    asm volatile("s_wait_asynccnt 1" ::: "memory");
}

// ---- block reduction: wave32 shuffle + LDS combine -------------------------
__device__ __forceinline__ float block_sum(float v, volatile float* red, int lane, int wid) {
#pragma unroll
    for (int off = 16; off > 0; off >>= 1) v += __shfl_xor(v, off, 32);
    if (lane == 0) red[wid] = v;
    __syncthreads();
    float c = 0.f;
#pragma unroll
    for (int i = 0; i < NWAVE; ++i) c += red[i];
    __syncthreads();
    return c;
}

// ---- emitT[b][t][s] = preds[b][lab_idx(s)][t] ------------------------------
__global__ __launch_bounds__(NTHR)
void emit_kernel(const float* __restrict__ preds, const int* __restrict__ labels,
                 const int* __restrict__ blankp, float* __restrict__ emitT) {
    const int t = blockIdx.x, b = blockIdx.y, s = threadIdx.x;
    if (s >= S_) return;
    const int blank = *blankp;
    const int lab = (s & 1) ? labels[b * L_ + ((s - 1) >> 1)] : blank;
    emitT[((size_t)b * T_ + t) * S_ + s] = preds[((size_t)b * C_ + lab) * T_ + t];
}

// ---- forward alpha scan ----------------------------------------------------
__global__ __launch_bounds__(NTHR)
void alpha_kernel(const float* __restrict__ emitT, const int* __restrict__ labels,
                  float* __restrict__ alphaT) {
    __shared__ float sh_a[NTHR + 2];     // [0..1] zero pads for the s-1 / s-2 shifts
    __shared__ float sh_e[2][NTHR];      // async double buffer for emit columns
    __shared__ float red[NWAVE];
    const int b = blockIdx.x;
    const int s = threadIdx.x;
    const int lane = s & 31, wid = s >> 5;
    const bool valid = s < S_;
    const int sc = valid ? s : (S_ - 1);     // clamp for idle lanes

    float m3 = 0.f;
    if (valid && (s & 1) && s >= 3) {
        int i = (s - 1) >> 1;
        m3 = (labels[b * L_ + i] != labels[b * L_ + i - 1]) ? 1.f : 0.f;
    }

    const float* eb = emitT + (size_t)b * T_ * S_;
    float* ab = alphaT + (size_t)b * T_ * S_;

    if (s < 2) sh_a[s] = 0.f;
    async_copy_b32(&sh_e[0][s], eb + sc);            // column t=0
    async_copy_b32(&sh_e[1][s], eb + S_ + sc);       // prefetch t=1
    wait_async_le1();                                // col 0 landed (in-order)
    float e = valid ? sh_e[0][s] : 0.f;
    float a = (s < 2) ? e : 0.f;                     // a0
    float c = block_sum(a, red, lane, wid);
    if (c > 0.f) a /= c;
    sh_a[2 + s] = a;
    if (valid) ab[s] = a;
    __syncthreads();

    for (int t = 1; t < T_; ++t) {
        const int tn = (t + 1 < T_) ? (t + 1) : (T_ - 1);
        async_copy_b32(&sh_e[(t + 1) & 1][s], eb + (size_t)tn * S_ + sc);
        wait_async_le1();                            // column t ready
        e = valid ? sh_e[t & 1][s] : 0.f;
        const float w = (s >= 2 * t - 767 && s < 2 * t + 2) ? 1.f : 0.f;
        const float an = (a + sh_a[2 + s - 1] + sh_a[s] * m3) * e * w;
        c = block_sum(an, red, lane, wid);           // barriers also fence sh_a reuse
        a = (c > 0.f) ? an / c : an;
        sh_a[2 + s] = a;
        if (valid) ab[(size_t)t * S_ + s] = a;
        __syncthreads();
    }
}

// ---- backward beta scan fused with lh / loss accumulation ------------------
__global__ __launch_bounds__(NTHR)
void beta_kernel(const float* __restrict__ emitT, const int* __restrict__ labels,
                 const float* __restrict__ alphaT, float* __restrict__ lossPartial) {
    __shared__ float sh_a[NTHR + 2];
    __shared__ float sh_e[2][NTHR];
    __shared__ float red[NWAVE];
    const int b = blockIdx.x;
    const int j = threadIdx.x;                 // reversed state index s'
    const int lane = j & 31, wid = j >> 5;
    const bool valid = j < S_;
    const int srev = valid ? (S_ - 1 - j) : 0; // memory index s = S-1-s'

    float m3 = 0.f;                            // mask3 of reversed labels
    if (valid && (j & 1) && j >= 3) {
        int i = (j - 1) >> 1;
        m3 = (labels[b * L_ + (L_ - 1 - i)] != labels[b * L_ + (L_ - i)]) ? 1.f : 0.f;
    }

    const float* eb = emitT + (size_t)b * T_ * S_;
    const float* ab = alphaT + (size_t)b * T_ * S_;

    if (j < 2) sh_a[j] = 0.f;
    async_copy_b32(&sh_e[0][j], eb + (size_t)(T_ - 1) * S_ + srev);  // t'=0 -> t=T-1
    async_copy_b32(&sh_e[1][j], eb + (size_t)(T_ - 2) * S_ + srev);  // prefetch
    wait_async_le1();
    float e = valid ? sh_e[0][j] : 0.f;
    float a = (j < 2) ? e : 0.f;
    float c = block_sum(a, red, lane, wid);
    if (c > 0.f) a /= c;
    float term = valid ? a * ab[(size_t)(T_ - 1) * S_ + srev] / e : 0.f;
    if (!valid) term = 0.f;
    float lh = block_sum(term, red, lane, wid);
    float loss = 0.f;
    if (j == 0) loss = -logf(lh);
    sh_a[2 + j] = a;
    __syncthreads();

    for (int tp = 1; tp < T_; ++tp) {
        const int t = T_ - 1 - tp;                       // real time index
        const int tnext = (t - 1 >= 0) ? (t - 1) : 0;
        async_copy_b32(&sh_e[(tp + 1) & 1][j], eb + (size_t)tnext * S_ + srev);
        wait_async_le1();
        e = valid ? sh_e[tp & 1][j] : 0.f;
        const float w = (j >= 2 * tp - 767 && j < 2 * tp + 2) ? 1.f : 0.f;
        const float an = (a + sh_a[2 + j - 1] + sh_a[j] * m3) * e * w;
        c = block_sum(an, red, lane, wid);
        a = (c > 0.f) ? an / c : an;
        term = valid ? a * ab[(size_t)t * S_ + srev] / e : 0.f;
        if (!valid) term = 0.f;
        lh = block_sum(term, red, lane, wid);
        if (j == 0) loss += -logf(lh);
        sh_a[2 + j] = a;
        __syncthreads();
    }
    if (j == 0) lossPartial[b] = loss;
}

// ---- batch mean via V_WMMA_F32_16X16X4_F32 (ones * losses) -----------------
__global__ void final_reduce(const float* __restrict__ lossPartial, float* __restrict__ out) {
    const int lane = threadIdx.x;        // 32 threads, one full wave, EXEC all ones
    const int n = lane & 15;
    const int khi = lane >> 4;           // 0 or 1
    v2f bm;                              // B: 4x16 fp32 holding the 64 losses
    bm.x = lossPartial[(0 + khi) * 16 + n];
    bm.y = lossPartial[(2 + khi) * 16 + n];
    v2f am; am.x = 1.f; am.y = 1.f;      // A: 16x4 all-ones
    v8f cm = {};
    v8f d = __builtin_amdgcn_wmma_f32_16x16x4_f32(
        false, am, false, bm, (short)0, cm, false, false);
    // every lane's d[0] is a column sum; summing all 32 lanes counts total twice
    float v = d[0];
#pragma unroll
    for (int off = 16; off > 0; off >>= 1) v += __shfl_xor(v, off, 32);
    if (lane == 0) out[0] = v * (1.0f / (2.0f * (float)B_));
}

extern "C" void kernel_launch(void* const* d_in, const int* in_sizes, int n_in,
                              void* d_out, int out_size, void* d_ws, size_t ws_size,
                              hipStream_t stream) {
    const float* preds  = (const float*)d_in[0];
    const int*   labels = (const int*)d_in[1];
    // d_in[2] = lengths (all == T, unused by the reference math)
    const int*   blankp = (const int*)d_in[3];

    float* ws     = (float*)d_ws;
    float* emitT  = ws;                                   // B*T*S
    float* alphaT = ws + (size_t)B_ * T_ * S_;            // B*T*S
    float* lossP  = alphaT + (size_t)B_ * T_ * S_;        // B
    float* out    = (float*)d_out;

    dim3 ge(T_, B_);
    emit_kernel<<<ge, NTHR, 0, stream>>>(preds, labels, blankp, emitT);
    alpha_kernel<<<B_, NTHR, 0, stream>>>(emitT, labels, alphaT);
    beta_kernel<<<B_, NTHR, 0, stream>>>(emitT, labels, alphaT, lossP);
    final_reduce<<<1, 32, 0, stream>>>(lossP, out);
}